// GraphNetGAT_83614423318872
// MI455X (gfx1250) — compile-verified
//
#include <hip/hip_runtime.h>
#include <float.h>
#include <math.h>

typedef float v2f __attribute__((ext_vector_type(2)));
typedef float v8f __attribute__((ext_vector_type(8)));

#define NEG_SLOPE 0.2f
#define BN_EPS    1e-5f
#define HEADS     4

// ---------------------------------------------------------------- utilities

__device__ inline void atomicMaxF(float* addr, float val) {
  unsigned int* ia = (unsigned int*)addr;
  unsigned int old = __builtin_bit_cast(unsigned int, *addr);
  while (true) {
    float f = __uint_as_float(old);
    if (f >= val) break;
    unsigned int assumed = old;
    old = atomicCAS(ia, assumed, __float_as_uint(val));
    if (old == assumed) break;
  }
}

__device__ inline void edge_sd(const int* __restrict__ ei, int Ereal, int e,
                               int& s, int& d) {
  if (e < Ereal) { s = ei[e]; d = ei[Ereal + e]; }
  else           { s = e - Ereal; d = s; }           // appended self-loops
}

__global__ void k_fill(float* __restrict__ p, float v, int n) {
  int i = blockIdx.x * blockDim.x + threadIdx.x;
  int stride = gridDim.x * blockDim.x;
  for (; i < n; i += stride) p[i] = v;
}

// ------------------------------------------------- WMMA fp32 GEMM (X @ W^T)
// Fast path: requires M%16==0, Nc%32==0, K even. One wave computes a 16x32
// output slab (two 16x16 tiles sharing the A operand), K stepped by 4 via
// V_WMMA_F32_16X16X4_F32. Main-loop loads are unpredicated global_load_b64.

__global__ void k_gemm_wmma_fast(const float* __restrict__ X,
                                 const float* __restrict__ W,
                                 float* __restrict__ OUT, int M, int K, int Nc) {
  const int lane = threadIdx.x & 31;
  const int wave = threadIdx.x >> 5;
  const int wpb  = blockDim.x >> 5;
  const int half = lane >> 4;              // 0: lanes 0-15, 1: lanes 16-31
  const int l16  = lane & 15;

  const int ntP  = Nc >> 5;                // pairs of 16-wide N tiles
  const int ntM  = M >> 4;
  const int tile = blockIdx.x * wpb + wave;
  if (tile >= ntM * ntP) return;           // wave-uniform: EXEC stays full
  const int m0 = (tile / ntP) << 4;
  const int n0 = (tile % ntP) << 5;

  const float* Arow  = X + (long)(m0 + l16) * K;        // A: row m0+l16
  const float* B0row = W + (long)(n0 + l16) * K;        // B tile 0 column
  const float* B1row = W + (long)(n0 + 16 + l16) * K;   // B tile 1 column

  v8f acc0 = {};
  v8f acc1 = {};
  const int Kmain = K & ~3;
  int ka = half << 1;                      // this lane's K pair within step
  for (int k0 = 0; k0 < Kmain; k0 += 4, ka += 4) {
    const v2f A  = *(const v2f*)(Arow  + ka);
    const v2f B0 = *(const v2f*)(B0row + ka);
    const v2f B1 = *(const v2f*)(B1row + ka);
    acc0 = __builtin_amdgcn_wmma_f32_16x16x4_f32(false, A, false, B0,
                                                 (short)0, acc0, false, false);
    acc1 = __builtin_amdgcn_wmma_f32_16x16x4_f32(false, A, false, B1,
                                                 (short)0, acc1, false, false);
  }
  if (Kmain < K) {                         // K%4==2 tail (layer 0, K=6)
    float a0 = 0.f, a1 = 0.f, b00 = 0.f, b01 = 0.f, b10 = 0.f, b11 = 0.f;
    if (half == 0) {                       // half==1 lanes: K indices OOB -> 0
      a0  = Arow[Kmain];  a1  = Arow[Kmain + 1];
      b00 = B0row[Kmain]; b01 = B0row[Kmain + 1];
      b10 = B1row[Kmain]; b11 = B1row[Kmain + 1];
    }
    const v2f A = {a0, a1};
    const v2f B0 = {b00, b01};
    const v2f B1 = {b10, b11};
    acc0 = __builtin_amdgcn_wmma_f32_16x16x4_f32(false, A, false, B0,
                                                 (short)0, acc0, false, false);
    acc1 = __builtin_amdgcn_wmma_f32_16x16x4_f32(false, A, false, B1,
                                                 (short)0, acc1, false, false);
  }
  // C/D layout: VGPR r -> row m0 + r + 8*half, col n0 + l16
  float* out0 = OUT + (long)(m0 + (half << 3)) * Nc + n0 + l16;
#pragma unroll
  for (int r = 0; r < 8; ++r) {
    out0[(long)r * Nc]      = acc0[r];
    out0[(long)r * Nc + 16] = acc1[r];
  }
}

// Generic fallback (bounds-checked), only used for unaligned shapes.
__global__ void k_gemm_wmma_safe(const float* __restrict__ X,
                                 const float* __restrict__ W,
                                 float* __restrict__ OUT, int M, int K, int Nc) {
  const int lane = threadIdx.x & 31;
  const int wave = threadIdx.x >> 5;
  const int wpb  = blockDim.x >> 5;
  const int half = lane >> 4;
  const int l16  = lane & 15;
  const int ntN = (Nc + 15) >> 4;
  const int ntM = (M + 15) >> 4;
  const int tile = blockIdx.x * wpb + wave;
  if (tile >= ntN * ntM) return;
  const int m0 = (tile / ntN) << 4;
  const int n0 = (tile % ntN) << 4;
  const int ra = m0 + l16;
  const int rb = n0 + l16;
  const bool aok = ra < M, bok = rb < Nc;
  v8f acc = {};
  for (int k0 = 0; k0 < K; k0 += 4) {
    const int ka = k0 + (half << 1);
    float a0 = 0.f, a1 = 0.f, b0 = 0.f, b1 = 0.f;
    if (aok && ka     < K) a0 = X[(long)ra * K + ka];
    if (aok && ka + 1 < K) a1 = X[(long)ra * K + ka + 1];
    if (bok && ka     < K) b0 = W[(long)rb * K + ka];
    if (bok && ka + 1 < K) b1 = W[(long)rb * K + ka + 1];
    const v2f A = {a0, a1};
    const v2f B = {b0, b1};
    acc = __builtin_amdgcn_wmma_f32_16x16x4_f32(false, A, false, B,
                                                (short)0, acc, false, false);
  }
  if (bok) {
#pragma unroll
    for (int r = 0; r < 8; ++r) {
      const int row = m0 + r + (half << 3);
      if (row < M) OUT[(long)row * Nc + n0 + l16] = acc[r];
    }
  }
}

// ------------------------------------------------------------- GAT kernels

// alpha_src/dst[n,h] = <xh[n,h,:], a_src/dst[h,:]>
__global__ void k_alpha(const float* __restrict__ xh,
                        const float* __restrict__ a_src,
                        const float* __restrict__ a_dst,
                        float* __restrict__ aS, float* __restrict__ aD,
                        int N, int C) {
  int i = blockIdx.x * blockDim.x + threadIdx.x;
  if (i >= N * HEADS) return;
  const int n = i >> 2, h = i & 3;
  const float* row = xh + (long)n * HEADS * C + (long)h * C;
  float s = 0.f, d = 0.f;
  for (int c = 0; c < C; ++c) {
    s += row[c] * a_src[h * C + c];
    d += row[c] * a_dst[h * C + c];
  }
  aS[i] = s;
  aD[i] = d;
}

// pass 1: leaky-relu logits + per-destination max
__global__ void k_edge1(const int* __restrict__ ei, int Ereal, int Etot,
                        const float* __restrict__ aS, const float* __restrict__ aD,
                        float* __restrict__ elog, float* __restrict__ segmax) {
  int i = blockIdx.x * blockDim.x + threadIdx.x;
  const int total = Etot * HEADS;
  const int stride = gridDim.x * blockDim.x;
  for (; i < total; i += stride) {
    const int e = i >> 2, h = i & 3;
    int s, d; edge_sd(ei, Ereal, e, s, d);
    float l = aS[(s << 2) + h] + aD[(d << 2) + h];
    l = l > 0.f ? l : NEG_SLOPE * l;
    elog[i] = l;
    atomicMaxF(&segmax[(d << 2) + h], l);
  }
}

// pass 2: exp(l - max) + per-destination sum
__global__ void k_edge2(const int* __restrict__ ei, int Ereal, int Etot,
                        const float* __restrict__ segmax,
                        float* __restrict__ elog, float* __restrict__ segsum) {
  int i = blockIdx.x * blockDim.x + threadIdx.x;
  const int total = Etot * HEADS;
  const int stride = gridDim.x * blockDim.x;
  for (; i < total; i += stride) {
    const int e = i >> 2, h = i & 3;
    int s, d; edge_sd(ei, Ereal, e, s, d);
    const float ev = expf(elog[i] - segmax[(d << 2) + h]);
    elog[i] = ev;
    atomicAdd(&segsum[(d << 2) + h], ev);
  }
}

// pass 3: weighted scatter, flattened over E*H*C for coalescing
__global__ void k_edge3(const int* __restrict__ ei, int Ereal, int Etot,
                        int C, const float* __restrict__ xh,
                        const float* __restrict__ elog,
                        const float* __restrict__ segsum,
                        float* __restrict__ out) {
  const int HC = HEADS * C;
  int i = blockIdx.x * blockDim.x + threadIdx.x;
  const int total = Etot * HC;
  const int stride = gridDim.x * blockDim.x;
  for (; i < total; i += stride) {
    const int e = i / HC;
    const int hc = i - e * HC;           // h*C + c
    const int h = hc / C;
    int s, d; edge_sd(ei, Ereal, e, s, d);
    const float w = elog[(e << 2) + h] / (segsum[(d << 2) + h] + 1e-16f);
    atomicAdd(&out[(long)d * HC + hc], xh[(long)s * HC + hc] * w);
  }
}

// ---------------------------------------------------------- BN / pool / FC

__global__ void k_bn_stats(const float* __restrict__ x, int N, int D,
                           float* __restrict__ stats) {
  __shared__ float s1[256];
  __shared__ float s2[256];
  const int c = blockIdx.x;
  float a = 0.f, b = 0.f;
  for (int n = threadIdx.x; n < N; n += blockDim.x) {
    const float v = x[(long)n * D + c];
    a += v; b += v * v;
  }
  s1[threadIdx.x] = a; s2[threadIdx.x] = b;
  __syncthreads();
  for (int off = blockDim.x >> 1; off > 0; off >>= 1) {
    if ((int)threadIdx.x < off) {
      s1[threadIdx.x] += s1[threadIdx.x + off];
      s2[threadIdx.x] += s2[threadIdx.x + off];
    }
    __syncthreads();
  }
  if (threadIdx.x == 0) {
    const float mu = s1[0] / N;
    stats[c]     = mu;
    stats[D + c] = s2[0] / N - mu * mu;
  }
}

__global__ void k_bn_relu(const float* __restrict__ x,
                          const float* __restrict__ stats,
                          const float* __restrict__ g,
                          const float* __restrict__ beta,
                          float* __restrict__ y, int N, int D) {
  int i = blockIdx.x * blockDim.x + threadIdx.x;
  const int total = N * D;
  const int stride = gridDim.x * blockDim.x;
  for (; i < total; i += stride) {
    const int c = i % D;
    const float v = (x[i] - stats[c]) * rsqrtf(stats[D + c] + BN_EPS) * g[c] + beta[c];
    y[i] = v > 0.f ? v : 0.f;
  }
}

__global__ void k_pool(const float* __restrict__ h, const int* __restrict__ batch,
                       float* __restrict__ pooled, int N, int D) {
  int i = blockIdx.x * blockDim.x + threadIdx.x;
  const int total = N * D;
  const int stride = gridDim.x * blockDim.x;
  for (; i < total; i += stride) {
    const int n = i / D, c = i - n * D;
    atomicMaxF(&pooled[(long)batch[n] * D + c], h[i]);
  }
}

__global__ void k_fc(const float* __restrict__ X, const float* __restrict__ W,
                     const float* __restrict__ bias, float* __restrict__ Y,
                     int R, int K, int J) {
  int i = blockIdx.x * blockDim.x + threadIdx.x;
  if (i >= R * J) return;
  const int r = i / J, j = i - (i / J) * J;
  float s = bias[j];
  for (int k = 0; k < K; ++k) s += X[(long)r * K + k] * W[(long)j * K + k];
  Y[i] = s;
}

__global__ void k_bn_rows_relu(float* __restrict__ X, const float* __restrict__ g,
                               const float* __restrict__ beta, int R, int D) {
  const int c = blockIdx.x * blockDim.x + threadIdx.x;
  if (c >= D) return;
  float a = 0.f, b = 0.f;
  for (int r = 0; r < R; ++r) { const float v = X[r * D + c]; a += v; b += v * v; }
  const float mu = a / R;
  const float var = b / R - mu * mu;
  const float sc = rsqrtf(var + BN_EPS) * g[c];
  for (int r = 0; r < R; ++r) {
    const float v = (X[r * D + c] - mu) * sc + beta[c];
    X[r * D + c] = v > 0.f ? v : 0.f;
  }
}

__global__ void k_l2(const float* __restrict__ X, float* __restrict__ out,
                     int D) {
  __shared__ float sm[64];
  const int r = blockIdx.x;
  const int c = threadIdx.x;
  const float v = X[r * D + c];
  sm[c] = v * v;
  __syncthreads();
  for (int off = blockDim.x >> 1; off > 0; off >>= 1) {
    if (c < off) sm[c] += sm[c + off];
    __syncthreads();
  }
  out[r * D + c] = v / fmaxf(sqrtf(sm[0]), 1e-12f);
}

// ------------------------------------------------------------------ driver

extern "C" void kernel_launch(void* const* d_in, const int* in_sizes, int n_in,
                              void* d_out, int out_size, void* d_ws, size_t ws_size,
                              hipStream_t stream) {
  (void)n_in; (void)out_size; (void)ws_size;

  const float* x     = (const float*)d_in[0];
  const int*   ei    = (const int*)d_in[1];
  const int*   batch = (const int*)d_in[2];
  const int N     = in_sizes[0] / 6;     // 20000
  const int Ereal = in_sizes[1] / 2;     // 320000
  const int Etot  = Ereal + N;           // + self loops
  const int G     = 32;

  // params in setup_inputs() dict insertion order
  const float *W[4], *aS_[4], *aD_[4], *bnG[4], *bnB[4];
  int idx = 3;
  for (int li = 0; li < 4; ++li) {
    W[li]   = (const float*)d_in[idx++];
    aS_[li] = (const float*)d_in[idx++];
    aD_[li] = (const float*)d_in[idx++];
    idx++;                               // b{li}: exactly cancelled by BN
    bnG[li] = (const float*)d_in[idx++];
    bnB[li] = (const float*)d_in[idx++];
  }
  const float *fcW[3], *fcb[3], *fbnG[2], *fbnB[2];
  for (int fi = 0; fi < 3; ++fi) {
    fcW[fi] = (const float*)d_in[idx++];
    fcb[fi] = (const float*)d_in[idx++];
    if (fi < 2) {
      fbnG[fi] = (const float*)d_in[idx++];
      fbnB[fi] = (const float*)d_in[idx++];
    }
  }

  static const int FIN[4] = {6, 32, 64, 128};
  static const int CC[4]  = {8, 16, 32, 256};

  // workspace carve-up (floats)
  float* ws = (float*)d_ws;
  size_t off = 0;
  auto alloc = [&](size_t n) { float* p = ws + off; off += n; return p; };
  float* xh     = alloc((size_t)N * 1024);
  float* agg    = alloc((size_t)N * 1024);
  float* hcur   = alloc((size_t)N * 1024);
  float* aS     = alloc((size_t)N * HEADS);
  float* aD     = alloc((size_t)N * HEADS);
  float* smax   = alloc((size_t)N * HEADS);
  float* ssum   = alloc((size_t)N * HEADS);
  float* elog   = alloc((size_t)Etot * HEADS);
  float* stats  = alloc(2 * 1024);
  float* pooled = alloc((size_t)G * 1024);
  float* f1     = alloc((size_t)G * 512);
  float* f2     = alloc((size_t)G * 256);
  float* f3     = alloc((size_t)G * 64);

  const float* hin = x;
  for (int li = 0; li < 4; ++li) {
    const int K = FIN[li], C = CC[li], HC = HEADS * C;

    // xh = hin @ W^T via WMMA fp32
    if ((N & 15) == 0 && (HC & 31) == 0 && (K & 1) == 0) {
      const int tiles = (N >> 4) * (HC >> 5);   // 16x32 slabs
      k_gemm_wmma_fast<<<(tiles + 3) / 4, 128, 0, stream>>>(hin, W[li], xh, N, K, HC);
    } else {
      const int tiles = ((N + 15) / 16) * ((HC + 15) / 16);
      k_gemm_wmma_safe<<<(tiles + 3) / 4, 128, 0, stream>>>(hin, W[li], xh, N, K, HC);
    }

    // per-node attention coefficients
    const int nh = N * HEADS;
    k_alpha<<<(nh + 255) / 256, 256, 0, stream>>>(xh, aS_[li], aD_[li], aS, aD, N, C);

    // softmax state + aggregation buffer init
    k_fill<<<256, 256, 0, stream>>>(smax, -FLT_MAX, nh);
    k_fill<<<256, 256, 0, stream>>>(ssum, 0.f, nh);
    k_fill<<<2048, 256, 0, stream>>>(agg, 0.f, N * HC);

    const int eh = Etot * HEADS;
    k_edge1<<<(eh + 255) / 256, 256, 0, stream>>>(ei, Ereal, Etot, aS, aD, elog, smax);
    k_edge2<<<(eh + 255) / 256, 256, 0, stream>>>(ei, Ereal, Etot, smax, elog, ssum);
    k_edge3<<<4096, 256, 0, stream>>>(ei, Ereal, Etot, C, xh, elog, ssum, agg);

    // BN + ReLU
    k_bn_stats<<<HC, 256, 0, stream>>>(agg, N, HC, stats);
    k_bn_relu<<<2048, 256, 0, stream>>>(agg, stats, bnG[li], bnB[li], hcur, N, HC);
    hin = hcur;
  }

  // global max pool over graphs
  k_fill<<<256, 256, 0, stream>>>(pooled, -FLT_MAX, G * 1024);
  k_pool<<<2048, 256, 0, stream>>>(hin, batch, pooled, N, 1024);

  // FC head
  k_fc<<<(G * 512 + 255) / 256, 256, 0, stream>>>(pooled, fcW[0], fcb[0], f1, G, 1024, 512);
  k_bn_rows_relu<<<(512 + 63) / 64, 64, 0, stream>>>(f1, fbnG[0], fbnB[0], G, 512);
  k_fc<<<(G * 256 + 255) / 256, 256, 0, stream>>>(f1, fcW[1], fcb[1], f2, G, 512, 256);
  k_bn_rows_relu<<<(256 + 63) / 64, 64, 0, stream>>>(f2, fbnG[1], fbnB[1], G, 256);
  k_fc<<<(G * 64 + 255) / 256, 256, 0, stream>>>(f2, fcW[2], fcb[2], f3, G, 256, 64);

  // row-wise L2 normalize -> output
  k_l2<<<G, 64, 0, stream>>>(f3, (float*)d_out, 64);
}